// DifferentiableMPC_28424093565699
// MI455X (gfx1250) — compile-verified
//
#include <hip/hip_runtime.h>

// ---------------------------------------------------------------------------
// Batched LQR-MPC (backward Riccati + forward rollout) for MI455X (gfx1250).
// One wave32 per batch element. All per-element state lives in LDS.
// Dense 16x16 products use V_WMMA_F32_16X16X4_F32.
// ---------------------------------------------------------------------------

#define T_STEPS 50
#define NS 12          // state dim
#define MC 4           // control dim
#define NSC 16         // NS + MC
#define WAVES_PER_BLOCK 2

typedef __attribute__((ext_vector_type(2))) float v2f;
typedef __attribute__((ext_vector_type(8))) float v8f;

// D = A(16x4) x B(4x16) + C(16x16), fp32 WMMA
__device__ __forceinline__ v8f wmma4(v2f a, v2f b, v8f c) {
  return __builtin_amdgcn_wmma_f32_16x16x4_f32(
      /*neg_a=*/false, a, /*neg_b=*/false, b,
      /*c_mod=*/(short)0, c, /*reuse_a=*/false, /*reuse_b=*/false);
}

// A-operand chunk: columns [kb, kb+4) of row-major 16x16 LDS matrix X.
// Layout (ISA 7.12.2): lanes 0-15 hold M=lane, VGPR0=K(kb), VGPR1=K(kb+1);
// lanes 16-31 hold K(kb+2), K(kb+3).
__device__ __forceinline__ v2f load_A_chunk(const float* X, int kb, int lane) {
  int hi = lane >> 4;
  int m  = lane & 15;
  int k0 = kb + (hi << 1);
  v2f r;
  r.x = X[m * 16 + k0];
  r.y = X[m * 16 + k0 + 1];
  return r;
}

// B-operand chunk: rows [kb, kb+4) of row-major 16x16 LDS matrix X.
// VGPR0: row kb (lanes 0-15) / row kb+2 (lanes 16-31); VGPR1: kb+1 / kb+3.
__device__ __forceinline__ v2f load_B_chunk(const float* X, int kb, int lane) {
  int hi = lane >> 4;
  int c  = lane & 15;
  int r0 = kb + (hi << 1);
  v2f r;
  r.x = X[r0 * 16 + c];
  r.y = X[(r0 + 1) * 16 + c];
  return r;
}

// K-hat operand from gains block g[n*4+m] (column-major 4x16, cols 13..15 = 0).
// Serves simultaneously as B-chunk of Kbig rows 12..15 and A-chunk of
// Kbig^T cols 12..15 (layouts coincide).
__device__ __forceinline__ v2f load_K_op(const float* g, int lane) {
  int hi = lane >> 4;
  int c  = lane & 15;
  int m0 = hi << 1;
  v2f r;
  r.x = g[c * 4 + m0];
  r.y = g[c * 4 + m0 + 1];
  return r;
}

// Store C/D-layout v8f into row-major 16x16 LDS matrix.
// VGPR r: row r (lanes 0-15) / row r+8 (lanes 16-31), col = lane&15.
__device__ __forceinline__ void store_D(float* X, v8f d, int lane) {
  int hi = lane >> 4;
  int c  = lane & 15;
  int rb = hi << 3;
#pragma unroll
  for (int r = 0; r < 8; ++r) X[(rb + r) * 16 + c] = d[r];
}

// Per-wave LDS footprint (floats)
#define OFF_F   0      // F = [A B; 0 0], 16x16 row-major
#define OFF_V   256    // V (padded 16x16)
#define OFF_W   512    // W = V*F, reused for G
#define OFF_QT  768    // Qt
#define OFF_Z   1024   // Z = Qt*Kbig
#define OFF_QTV 1280   // qt vector (16)
#define OFF_VV  1296   // v vector (16)
#define OFF_XU  1312   // [x; u] (16)
#define OFF_G   1344   // gains: T * 64 floats (K-hat col-major, zero padded)
#define PW      (1344 + T_STEPS * 64)   // 4544 floats per wave

__global__ __launch_bounds__(32 * WAVES_PER_BLOCK)
void mpc_lqr_kernel(const float* __restrict__ x_init,
                    const float* __restrict__ Qd,
                    const float* __restrict__ p,
                    const float* __restrict__ Amat,
                    const float* __restrict__ Bmat,
                    float* __restrict__ out, int nb) {
  __shared__ float smem[WAVES_PER_BLOCK * PW];
  const int lane = threadIdx.x & 31;
  const int wave = threadIdx.x >> 5;
  const int b = blockIdx.x * WAVES_PER_BLOCK + wave;
  if (b >= nb) return;

  float* S   = smem + wave * PW;
  float* F   = S + OFF_F;
  float* V   = S + OFF_V;
  float* W   = S + OFF_W;
  float* Qt  = S + OFF_QT;
  float* Z   = S + OFF_Z;
  float* qtv = S + OFF_QTV;
  float* vv  = S + OFF_VV;
  float* xu  = S + OFF_XU;
  float* gains = S + OFF_G;

  const int hi  = lane >> 4;
  const int col = lane & 15;

  // ---- Init F = [A | B] padded to 16x16, V = 0, v = 0 (per-wave copies) ----
  for (int idx = lane; idx < 256; idx += 32) {
    int i = idx >> 4, j = idx & 15;
    float val = 0.f;
    if (i < NS) val = (j < NS) ? Amat[i * NS + j] : Bmat[i * MC + (j - NS)];
    F[idx] = val;
    V[idx] = 0.f;
  }
  if (hi == 0) vv[col] = 0.f;

  // Constant WMMA operands: fF[k] = B-chunk of F rows 4k..4k+3
  //                               = A-chunk of F^T cols 4k..4k+3 (same regs).
  v2f fF0 = load_B_chunk(F, 0, lane);
  v2f fF1 = load_B_chunk(F, 4, lane);
  v2f fF2 = load_B_chunk(F, 8, lane);

  // C = diag(Q[b]) in D layout; p[b] per-lane.
  const float ql = Qd[b * NSC + col];
  const float pl = p[b * NSC + col];
  v8f qC;
#pragma unroll
  for (int r = 0; r < 8; ++r) {
    int row = (hi << 3) + r;
    qC[r] = (row == col) ? ql : 0.f;
  }

  // ======================= Backward Riccati recursion ======================
#pragma unroll 1
  for (int it = 0; it < T_STEPS; ++it) {
    // W = V * F   (V cols 12..15 are zero -> 3 K-chunks)
    v8f w = {};
    w = wmma4(load_A_chunk(V, 0, lane), fF0, w);
    w = wmma4(load_A_chunk(V, 4, lane), fF1, w);
    w = wmma4(load_A_chunk(V, 8, lane), fF2, w);
    store_D(W, w, lane);

    // Qt = F^T * W + diag(Q)   (F rows 12..15 zero -> 3 K-chunks)
    v8f q = qC;
    q = wmma4(fF0, load_B_chunk(W, 0, lane), q);
    q = wmma4(fF1, load_B_chunk(W, 4, lane), q);
    q = wmma4(fF2, load_B_chunk(W, 8, lane), q);
    store_D(Qt, q, lane);

    // qt = p + F^T v   (lane a = col; lanes 0..15 store)
    float qa = pl;
#pragma unroll
    for (int i = 0; i < NS; ++i) qa += F[i * 16 + col] * vv[i];
    if (hi == 0) qtv[col] = qa;

    // ---- 4x4 SPD solve: sol = -Quu^{-1} [Qux | qu], lane n owns column n ---
    const int n = col;
    float M4[4][4];
#pragma unroll
    for (int i = 0; i < 4; ++i)
#pragma unroll
      for (int j = 0; j < 4; ++j) M4[i][j] = Qt[(12 + i) * 16 + 12 + j];
    float r4[4];
#pragma unroll
    for (int i = 0; i < 4; ++i)
      r4[i] = (n < NS) ? Qt[(12 + i) * 16 + n]
                       : ((n == NS) ? qtv[12 + i] : 0.f);
    // unpivoted Gaussian elimination (Quu SPD)
#pragma unroll
    for (int kk = 0; kk < 4; ++kk) {
      float inv = 1.f / M4[kk][kk];
#pragma unroll
      for (int i = kk + 1; i < 4; ++i) {
        float f = M4[i][kk] * inv;
#pragma unroll
        for (int j = kk + 1; j < 4; ++j) M4[i][j] -= f * M4[kk][j];
        r4[i] -= f * r4[kk];
      }
    }
    float s3 = r4[3] / M4[3][3];
    float s2 = (r4[2] - M4[2][3] * s3) / M4[2][2];
    float s1 = (r4[1] - M4[1][2] * s2 - M4[1][3] * s3) / M4[1][1];
    float s0 = (r4[0] - M4[0][1] * s1 - M4[0][2] * s2 - M4[0][3] * s3) / M4[0][0];
    float s[4] = {-s0, -s1, -s2, -s3};

    // store K-hat column (col 12 = kff, cols 13..15 zero), time-reversed slot
    float* gt = gains + (T_STEPS - 1 - it) * 64;
    if (hi == 0) {
      gt[n * 4 + 0] = s[0];
      gt[n * 4 + 1] = s[1];
      gt[n * 4 + 2] = s[2];
      gt[n * 4 + 3] = s[3];
    }

    // Z = Qt * Kbig-hat  (only rows 12..15 of Kbig nonzero -> 1 WMMA)
    v2f kop = load_K_op(gt, lane);
    v8f zz = {};
    v8f z = wmma4(load_A_chunk(Qt, 12, lane), kop, zz);
    store_D(Z, z, lane);

    // G = Kbig^T * Z  (1 WMMA); G = K^T Quu K, col 12 = K^T Quu kff
    v8f gz = {};
    v8f g = wmma4(kop, load_B_chunk(Z, 12, lane), gz);
    store_D(W, g, lane);  // reuse W as G scratch

    // V+ = mask_{12x12}( Qt + Z + Z^T + G )
    const bool colok = (col < NS);
    v8f vn;
#pragma unroll
    for (int r = 0; r < 8; ++r) {
      int row = (hi << 3) + r;
      float zt = Z[col * 16 + row];
      float val = q[r] + z[r] + zt + g[r];
      vn[r] = (colok && row < NS) ? val : 0.f;
    }
    store_D(V, vn, lane);

    // v+ = qx + Qxu kff + K^T Quu kff + K^T qu
    if (hi == 0 && n < NS) {
      float acc = qtv[n] + Z[n * 16 + 12] + W[n * 16 + 12];
      acc += s[0] * qtv[12] + s[1] * qtv[13] + s[2] * qtv[14] + s[3] * qtv[15];
      vv[n] = acc;
    }
  }

  // =========================== Forward rollout ============================
  if (hi == 0) xu[col] = (col < NS) ? x_init[b * NS + col] : 0.f;

#pragma unroll 1
  for (int t = 0; t < T_STEPS; ++t) {
    const float* gt = gains + t * 64;
    // u[m] = sum_n K[m][n] x[n] + kff[m]   (lanes 0..3)
    if (lane < MC) {
      float u = gt[48 + lane];
#pragma unroll
      for (int nn = 0; nn < NS; ++nn) u += gt[nn * 4 + lane] * xu[nn];
      out[(t * nb + b) * MC + lane] = u;
      xu[NS + lane] = u;
    }
    // x+[i] = F[i,:] . [x; u]   (lanes 0..11)
    float xn = 0.f;
    if (lane < NS) {
#pragma unroll
      for (int a = 0; a < NSC; ++a) xn += F[lane * 16 + a] * xu[a];
    }
    if (lane < NS) xu[lane] = xn;
  }
}

extern "C" void kernel_launch(void* const* d_in, const int* in_sizes, int n_in,
                              void* d_out, int out_size, void* d_ws, size_t ws_size,
                              hipStream_t stream) {
  (void)n_in; (void)d_ws; (void)ws_size; (void)out_size;
  const float* x_init = (const float*)d_in[0];
  const float* Qd     = (const float*)d_in[1];
  const float* p      = (const float*)d_in[2];
  const float* Amat   = (const float*)d_in[3];
  const float* Bmat   = (const float*)d_in[4];
  float* out = (float*)d_out;
  const int nb = in_sizes[0] / NS;  // 8192
  const int blocks = (nb + WAVES_PER_BLOCK - 1) / WAVES_PER_BLOCK;
  mpc_lqr_kernel<<<blocks, 32 * WAVES_PER_BLOCK, 0, stream>>>(
      x_init, Qd, p, Amat, Bmat, out, nb);
}